// linear_model_13374528159965
// MI455X (gfx1250) — compile-verified
//
#include <hip/hip_runtime.h>
#include <hip/hip_bf16.h>
#include <float.h>

// ---------------------------------------------------------------------------
// Problem constants (match reference)
// ---------------------------------------------------------------------------
#define BT      16384      // B*T tokens
#define H       256
#define NUM_CAT 8
#define NUM_CONT 24
#define NUM_FEAT 32
#define OUT_DIM 32
#define VPAD    513
#define NEG_INF (-1e30f)
#define APAD    260        // LDS row stride (floats): 16B-aligned rows, banks spread 4/row

__device__ __constant__ int g_vocabs[NUM_CAT] = {512, 512, 256, 256, 128, 128, 64, 64};

typedef __attribute__((ext_vector_type(2))) float v2f;
typedef __attribute__((ext_vector_type(8))) float v8f;

// Cooperatively load a 16xH f32 tile (rows tbase..tbase+15 of src) into padded LDS.
// 256 threads, 4 x float4 each; ds_store_b128, conflict-free padded layout.
__device__ __forceinline__ void load_tile_lds(float (*As)[APAD], const float* __restrict__ src,
                                              int tbase) {
#pragma unroll
  for (int i = 0; i < 4; ++i) {
    int flat = threadIdx.x + i * 256;            // float4 index 0..1023
    int m  = flat >> 6;                          // 64 float4 per row
    int c4 = flat & 63;
    float4 v = reinterpret_cast<const float4*>(src + (size_t)(tbase + m) * H)[c4];
    *reinterpret_cast<float4*>(&As[m][c4 * 4]) = v;
  }
}

// ---------------------------------------------------------------------------
// Kernel 1: h0[token][h] = mean_f(embs[f][id_f][h]) + x_cont @ feat_W + feat_b
// ---------------------------------------------------------------------------
__global__ void k_h0(const float* __restrict__ x, const float* __restrict__ embs,
                     const float* __restrict__ feat_W, const float* __restrict__ feat_b,
                     float* __restrict__ h0) {
  const int token = blockIdx.x;
  const int hh = threadIdx.x;
  __shared__ float xs[NUM_FEAT];
  if (hh < NUM_FEAT) xs[hh] = x[token * NUM_FEAT + hh];
  __syncthreads();

  float acc_emb = 0.0f;
#pragma unroll
  for (int f = 0; f < NUM_CAT; ++f) {
    int id = (int)xs[f] + 1;                     // ids stored as float(id-1)
    acc_emb += embs[((size_t)f * VPAD + id) * H + hh];
  }
  float acc = feat_b[hh] + acc_emb * (1.0f / NUM_CAT);
#pragma unroll 4
  for (int j = 0; j < NUM_CONT; ++j)
    acc += xs[NUM_CAT + j] * feat_W[j * H + hh];
  h0[(size_t)token * H + hh] = acc;
}

// ---------------------------------------------------------------------------
// Kernel 2: h = h0 @ model_W + model_b via V_WMMA_F32_16X16X4_F32.
// One block per 16-token M-tile; A tile staged in LDS once; 8 waves cover the
// 16 N-tiles (2 each). K=256 -> 64 WMMA steps per N-tile.
// A 16x4 layout: lanes 0-15 = M rows, K=0,1; lanes 16-31 = K=2,3.
// B 4x16 layout: lanes 0-15 = N cols, K=0,1; lanes 16-31 = K=2,3.
// ---------------------------------------------------------------------------
__global__ void k_model(const float* __restrict__ h0, const float* __restrict__ model_W,
                        const float* __restrict__ model_b, float* __restrict__ h) {
  __shared__ float As[16][APAD];
  const int mtile = blockIdx.x;                  // 0..1023
  const int tbase = mtile * 16;
  const int lane  = threadIdx.x & 31;
  const int wid   = threadIdx.x >> 5;            // 0..7
  const int l16   = lane & 15;
  const int kp    = (lane >> 4) * 2;             // 0 or 2

  load_tile_lds(As, h0, tbase);
  __syncthreads();

  const float* Arow = &As[l16][kp];              // LDS: row l16, K pair
#pragma unroll
  for (int nt = 0; nt < 2; ++nt) {
    const int nbase = (wid * 2 + nt) * 16;
    const float* Bp = model_W + (size_t)kp * H + (nbase + l16);

    v8f acc = {};
    for (int k0 = 0; k0 < H; k0 += 4) {
      v2f a; a.x = Arow[k0];           a.y = Arow[k0 + 1];       // ds_load_b64
      v2f b; b.x = Bp[(size_t)k0 * H]; b.y = Bp[(size_t)(k0 + 1) * H];
      acc = __builtin_amdgcn_wmma_f32_16x16x4_f32(false, a, false, b,
                                                  (short)0, acc, false, false);
    }
    const float bias = model_b[nbase + l16];
#pragma unroll
    for (int r = 0; r < 8; ++r) {
      int m = r + ((lane >> 4) << 3);            // C layout: VGPR r -> M=r / r+8
      h[(size_t)(tbase + m) * H + nbase + l16] = acc[r] + bias;
    }
  }
}

// ---------------------------------------------------------------------------
// Kernel 3: u = h @ fcu_W + fcu_b ; o = h @ fco_W + fco_b (u cols 0..7 are
// overwritten later by k_logits). 64 threads per token: 0-31 -> u, 32-63 -> o.
// ---------------------------------------------------------------------------
__global__ void k_uo(const float* __restrict__ h,
                     const float* __restrict__ fcu_W, const float* __restrict__ fcu_b,
                     const float* __restrict__ fco_W, const float* __restrict__ fco_b,
                     float* __restrict__ out_u, float* __restrict__ out_o) {
  const int token = blockIdx.x;
  __shared__ float hs[H];
  for (int i = threadIdx.x; i < H; i += 64) hs[i] = h[(size_t)token * H + i];
  __syncthreads();

  const int col = threadIdx.x & 31;
  const bool is_o = threadIdx.x >= 32;
  const float* W = is_o ? fco_W : fcu_W;
  float s = is_o ? fco_b[col] : fcu_b[col];
#pragma unroll 8
  for (int k = 0; k < H; ++k) s += hs[k] * W[k * OUT_DIM + col];
  (is_o ? out_o : out_u)[(size_t)token * OUT_DIM + col] = s;
}

// ---------------------------------------------------------------------------
// Kernel 4: c[f,token,v] = masked(h @ cat_W[f] + cat_b[f]); argmax -> u[:,:,f].
// One block per 16-token tile; its 8 waves = the 8 category heads f, all
// sharing one LDS-resident A tile (8x global-traffic cut). Each wave: 33
// N-tiles x 64 K-step WMMAs; running per-lane (max,idx) across ascending n
// (first-index ties), then 4-step shfl_xor reduction per 16-lane half.
// ---------------------------------------------------------------------------
__global__ void k_logits(const float* __restrict__ h, const float* __restrict__ cat_W,
                         const float* __restrict__ cat_b,
                         float* __restrict__ out_u, float* __restrict__ out_c) {
  __shared__ float As[16][APAD];
  const int mtile = blockIdx.x;                  // 0..1023
  const int tbase = mtile * 16;
  const int lane  = threadIdx.x & 31;
  const int f     = threadIdx.x >> 5;            // wave id == category head
  const int l16   = lane & 15;
  const int kp    = (lane >> 4) * 2;
  const int vocab1 = g_vocabs[f] + 1;            // # valid vocab entries

  load_tile_lds(As, h, tbase);
  __syncthreads();

  const float* Arow = &As[l16][kp];
  const float* Bf = cat_W + (size_t)f * H * VPAD;
  const float* bb = cat_b + (size_t)f * VPAD;
  float* cbase    = out_c + (size_t)(f * BT + tbase) * VPAD;

  float best[8];
  int   bidx[8];
#pragma unroll
  for (int r = 0; r < 8; ++r) { best[r] = -FLT_MAX; bidx[r] = 0; }

  const int NTILES = (VPAD + 15) / 16;           // 33 (last tile partial)
  for (int ntile = 0; ntile < NTILES; ++ntile) {
    const int nbase = ntile * 16;
    const int n  = nbase + l16;
    const int nc = (n < VPAD) ? n : (VPAD - 1);  // clamp B loads in bounds

    v8f acc = {};
    for (int k0 = 0; k0 < H; k0 += 4) {
      v2f a; a.x = Arow[k0]; a.y = Arow[k0 + 1];               // ds_load_b64
      v2f b;
      b.x = Bf[(size_t)(k0 + kp) * VPAD + nc];
      b.y = Bf[(size_t)(k0 + kp + 1) * VPAD + nc];
      acc = __builtin_amdgcn_wmma_f32_16x16x4_f32(false, a, false, b,
                                                  (short)0, acc, false, false);
    }

    const float biasc = (n < VPAD) ? bb[nc] : 0.0f;
    const bool  valid = (n < vocab1);            // vocab1 <= VPAD always
#pragma unroll
    for (int r = 0; r < 8; ++r) {
      int m = r + ((lane >> 4) << 3);
      float cv = valid ? (acc[r] + biasc) : NEG_INF;
      if (cv > best[r]) { best[r] = cv; bidx[r] = n; }   // strict > keeps lowest n
      if (n < VPAD) cbase[(size_t)m * VPAD + n] = cv;
    }
  }

  // cross-lane argmax within each 16-lane half (per C/D row)
#pragma unroll
  for (int r = 0; r < 8; ++r) {
    float bv = best[r];
    int   bi = bidx[r];
#pragma unroll
    for (int off = 8; off >= 1; off >>= 1) {
      float ov = __shfl_xor(bv, off, 32);
      int   oi = __shfl_xor(bi, off, 32);
      if (ov > bv || (ov == bv && oi < bi)) { bv = ov; bi = oi; }
    }
    if (l16 == 0) {
      int m = r + ((lane >> 4) << 3);
      out_u[(size_t)(tbase + m) * OUT_DIM + f] = (float)bi - 1.0f;
    }
  }
}

// ---------------------------------------------------------------------------
// Launch
// ---------------------------------------------------------------------------
extern "C" void kernel_launch(void* const* d_in, const int* in_sizes, int n_in,
                              void* d_out, int out_size, void* d_ws, size_t ws_size,
                              hipStream_t stream) {
  const float* x       = (const float*)d_in[0];
  const float* embs    = (const float*)d_in[1];
  const float* feat_W  = (const float*)d_in[2];
  const float* feat_b  = (const float*)d_in[3];
  const float* model_W = (const float*)d_in[4];
  const float* model_b = (const float*)d_in[5];
  const float* fcu_W   = (const float*)d_in[6];
  const float* fcu_b   = (const float*)d_in[7];
  const float* fco_W   = (const float*)d_in[8];
  const float* fco_b   = (const float*)d_in[9];
  const float* cat_W   = (const float*)d_in[10];
  const float* cat_b   = (const float*)d_in[11];

  float* out   = (float*)d_out;
  float* out_u = out;                            // (B,T,32)
  float* out_o = out + (size_t)BT * OUT_DIM;     // (B,T,32)
  float* out_c = out_o + (size_t)BT * OUT_DIM;   // (8,B,T,513)

  float* h0 = (float*)d_ws;                      // 16 MB
  float* hh = h0 + (size_t)BT * H;               // 16 MB

  // 1) embedding gather + feature GEMM -> h0
  k_h0<<<BT, 256, 0, stream>>>(x, embs, feat_W, feat_b, h0);
  // 2) h = h0 @ model_W + b  (1 block per M-tile, A tile in LDS)
  k_model<<<BT / 16, 256, 0, stream>>>(h0, model_W, model_b, hh);
  // 3) u, o heads
  k_uo<<<BT, 64, 0, stream>>>(hh, fcu_W, fcu_b, fco_W, fco_b, out_u, out_o);
  // 4) logits + mask + argmax (1 block per M-tile, 8 waves = 8 heads, shared A)
  k_logits<<<BT / 16, 256, 0, stream>>>(hh, cat_W, cat_b, out_u, out_c);
}